// LinearLoss_18769007084481
// MI455X (gfx1250) — compile-verified
//
#include <hip/hip_runtime.h>
#include <math.h>

// ---------------------------------------------------------------------------
// Center-loss scalar: two streaming passes over x (2 x 205 MB -> ~18us @23.3TB/s)
// Pass 1 does the class-centroid segment-sum as a one-hot GEMM on
// V_WMMA_F32_16X16X4_F32 (CDNA5 wave32 WMMA).
// ---------------------------------------------------------------------------

typedef float v8f __attribute__((ext_vector_type(8)));
typedef float v2f __attribute__((ext_vector_type(2)));

#define CL_D       256
#define CL_C       5
#define CL_TROWS   16            // rows per tile (N = 200000 is divisible by 16)
#define CL_LSTRIDE 260           // padded LDS row stride (floats)
#define EPS_F      1e-6f

// workspace layout (floats)
#define WS_SUMS    0             // [5*256]
#define WS_COUNTS  1280          // [5]
#define WS_CENTER  1536          // [5*256]
#define WS_DSUM    2816          // [5]
#define WS_ZERO_N  2821

// ---------------------------------------------------------------------------
__global__ void cl_zero(float* ws, int n) {
    int i = blockIdx.x * blockDim.x + threadIdx.x;
    if (i < n) ws[i] = 0.0f;
}

// ---------------------------------------------------------------------------
// Pass 1: per-class counts + centroid sums via WMMA one-hot GEMM.
// Block = 512 threads = 16 waves; wave w owns feature slice [w*16, w*16+16).
// Each tile: 16 rows x 256 cols of x staged into LDS, 4 WMMAs per wave
// (sample groups of K=4).
__global__ __launch_bounds__(512) void cl_sums(const float* __restrict__ x,
                                               const float* __restrict__ y,
                                               float* __restrict__ gsums,
                                               float* __restrict__ gcounts,
                                               int ntiles) {
    __shared__ __attribute__((aligned(16))) float ldsx[CL_TROWS * CL_LSTRIDE];
    __shared__ int ldslab[CL_TROWS];
    __shared__ int ldscnt[CL_C];

    const int tid   = threadIdx.x;
    const int lane  = tid & 31;
    const int wave  = tid >> 5;
    const int mlane = lane & 15;   // M for A-frag, N for B-frag, N for D-frag
    const int khalf = lane >> 4;   // selects K pair {0,1} vs {2,3}
    const int fb    = wave * 16;   // feature base of this wave's slice

    if (tid < CL_C) ldscnt[tid] = 0;

    v8f acc = {};

    for (int t = blockIdx.x; t < ntiles; t += gridDim.x) {
        __syncthreads();  // previous tile fully consumed before overwrite

        const long base = (long)t * (CL_TROWS * CL_D);
        const float4* src = (const float4*)(x + base);

        // prefetch next tile this block will touch (global_prefetch_b8 path)
        {
            long nb = (long)(t + gridDim.x) * (CL_TROWS * CL_D);
            if (t + gridDim.x < ntiles && tid < 128)
                __builtin_prefetch(x + nb + (long)tid * 32, 0, 3);
        }

        // coalesced tile load: 1024 float4s, 2 per thread
        #pragma unroll
        for (int q = 0; q < 2; ++q) {
            int idx = tid + q * 512;                 // 0..1023
            int r   = idx >> 6;                      // row (64 float4 per row)
            int c4  = idx & 63;
            float4 v4 = src[idx];
            *(float4*)&ldsx[r * CL_LSTRIDE + c4 * 4] = v4;
        }

        // labels for the 16 rows of this tile (argmax, first-max-wins)
        if (tid < CL_TROWS) {
            const float* yr = y + (long)(t * CL_TROWS + tid) * CL_C;
            float best = yr[0]; int lab = 0;
            #pragma unroll
            for (int c = 1; c < CL_C; ++c) {
                float v = yr[c];
                if (v > best) { best = v; lab = c; }
            }
            ldslab[tid] = lab;
            atomicAdd(&ldscnt[lab], 1);
        }
        __syncthreads();

        // 4 sample-groups of K=4 -> 4 WMMAs; EXEC is all-1s here (uniform flow)
        #pragma unroll
        for (int g = 0; g < 4; ++g) {
            const int r0 = g * 4 + khalf * 2;        // K rows handled by this half
            const int l0 = ldslab[r0];
            const int l1 = ldslab[r0 + 1];
            v2f a, b;
            a.x = (l0 == mlane) ? 1.0f : 0.0f;       // A[m][k]   one-hot
            a.y = (l1 == mlane) ? 1.0f : 0.0f;       // A[m][k+1]
            b.x = ldsx[(r0    ) * CL_LSTRIDE + fb + mlane];  // B[k][n]
            b.y = ldsx[(r0 + 1) * CL_LSTRIDE + fb + mlane];  // B[k+1][n]
#if __has_builtin(__builtin_amdgcn_wmma_f32_16x16x4_f32)
            acc = __builtin_amdgcn_wmma_f32_16x16x4_f32(
                false, a, false, b, (short)0, acc, false, false);
#else
            // layout-identical fallback: D[m][n] += sum_k A[m][k]*B[k][n],
            // lane holds D[m = j + khalf*8][n = mlane] in acc[j]
            if (khalf == 0) {
                #pragma unroll
                for (int j = 0; j < CL_C; ++j) {
                    #pragma unroll
                    for (int kk = 0; kk < 4; ++kk) {
                        int rr = g * 4 + kk;
                        if (ldslab[rr] == j) acc[j] += ldsx[rr * CL_LSTRIDE + fb + mlane];
                    }
                }
            }
            (void)a; (void)b;
#endif
        }
    }

    __syncthreads();
    // classes 0..4 live in acc[0..4] of lanes 0..15 (M = vgpr + khalf*8)
    if (lane < 16) {
        #pragma unroll
        for (int j = 0; j < CL_C; ++j)
            atomicAdd(&gsums[j * CL_D + fb + mlane], acc[j]);
    }
    if (tid < CL_C) atomicAdd(&gcounts[tid], (float)ldscnt[tid]);
}

// ---------------------------------------------------------------------------
// centers[c][d] = counts[c] > 0 ? sums/counts : 0
__global__ void cl_center(const float* __restrict__ gsums,
                          const float* __restrict__ gcounts,
                          float* __restrict__ center) {
    int i = blockIdx.x * blockDim.x + threadIdx.x;
    if (i < CL_C * CL_D) {
        float cnt = gcounts[i >> 8];
        center[i] = (cnt > 0.0f) ? gsums[i] / cnt : 0.0f;
    }
}

// ---------------------------------------------------------------------------
// Pass 2: one wave per row; d = ||x - center[label] + eps||, segment-sum.
__global__ __launch_bounds__(256) void cl_dist(const float* __restrict__ x,
                                               const float* __restrict__ y,
                                               const float* __restrict__ center,
                                               float* __restrict__ gdsum,
                                               int N) {
    __shared__ __attribute__((aligned(16))) float c_s[CL_C * CL_D];
    __shared__ float dsum_s[CL_C];

    const int tid = threadIdx.x;
    for (int i = tid; i < CL_C * CL_D; i += blockDim.x) c_s[i] = center[i];
    if (tid < CL_C) dsum_s[tid] = 0.0f;
    __syncthreads();

    const int lane   = tid & 31;
    const int wave   = tid >> 5;
    const int gwave  = blockIdx.x * (blockDim.x >> 5) + wave;
    const int stride = gridDim.x * (blockDim.x >> 5);

    for (int r = gwave; r < N; r += stride) {
        const float* yr = y + (long)r * CL_C;
        float best = yr[0]; int lab = 0;
        #pragma unroll
        for (int c = 1; c < CL_C; ++c) {
            float v = yr[c];
            if (v > best) { best = v; lab = c; }
        }

        const float4* xr = (const float4*)(x + (long)r * CL_D);
        float s = 0.0f;
        #pragma unroll
        for (int h = 0; h < 2; ++h) {
            int c4 = h * 32 + lane;                   // float4 index 0..63
            float4 xv = xr[c4];
            float4 cv = *(const float4*)&c_s[lab * CL_D + c4 * 4];
            float t0 = xv.x - cv.x + EPS_F; s = fmaf(t0, t0, s);
            float t1 = xv.y - cv.y + EPS_F; s = fmaf(t1, t1, s);
            float t2 = xv.z - cv.z + EPS_F; s = fmaf(t2, t2, s);
            float t3 = xv.w - cv.w + EPS_F; s = fmaf(t3, t3, s);
        }
        #pragma unroll
        for (int off = 16; off > 0; off >>= 1) s += __shfl_xor(s, off, 32);
        if (lane == 0) atomicAdd(&dsum_s[lab], sqrtf(s));
    }

    __syncthreads();
    if (tid < CL_C) atomicAdd(&gdsum[tid], dsum_s[tid]);
}

// ---------------------------------------------------------------------------
// Final scalar on one wave: present mask, loss_pos, pairwise center distances.
__global__ void cl_final(const float* __restrict__ center,
                         const float* __restrict__ gcounts,
                         const float* __restrict__ gdsum,
                         float* __restrict__ out) {
    const int lane = threadIdx.x;  // 32 threads

    float counts[CL_C];
    bool  pres[CL_C];
    float real_n = 0.0f;

    #pragma unroll
    for (int c = 0; c < CL_C; ++c) {
        counts[c] = gcounts[c];
        float s = 0.0f;
        #pragma unroll
        for (int k = 0; k < 8; ++k) s += center[c * CL_D + lane + k * 32];
        #pragma unroll
        for (int off = 16; off > 0; off >>= 1) s += __shfl_xor(s, off, 32);
        pres[c] = (counts[c] > 0.0f) && (s != 0.0f);
        if (pres[c]) real_n += 1.0f;
    }

    float loss_pos = 0.0f;
    #pragma unroll
    for (int c = 0; c < CL_C; ++c)
        if (pres[c]) loss_pos += gdsum[c] / fmaxf(counts[c], 1.0f);

    float neg_sum = 0.0f;
    #pragma unroll
    for (int c1 = 0; c1 < CL_C; ++c1) {
        #pragma unroll
        for (int c2 = c1 + 1; c2 < CL_C; ++c2) {
            if (pres[c1] && pres[c2]) {   // uniform across the wave
                float s = 0.0f;
                #pragma unroll
                for (int k = 0; k < 8; ++k) {
                    float d = center[c1 * CL_D + lane + k * 32]
                            - center[c2 * CL_D + lane + k * 32] + EPS_F;
                    s = fmaf(d, d, s);
                }
                #pragma unroll
                for (int off = 16; off > 0; off >>= 1) s += __shfl_xor(s, off, 32);
                neg_sum += sqrtf(s);
            }
        }
    }

    float n_pairs  = real_n * (real_n - 1.0f) * 0.5f;
    float loss_neg = (real_n > 1.0f) ? neg_sum / fmaxf(n_pairs, 1.0f) : 0.0f;
    loss_pos = (real_n > 0.0f) ? loss_pos / fmaxf(real_n, 1.0f) : 1.0f;

    if (lane == 0) out[0] = loss_pos / (loss_neg + loss_pos);
}

// ---------------------------------------------------------------------------
extern "C" void kernel_launch(void* const* d_in, const int* in_sizes, int n_in,
                              void* d_out, int out_size, void* d_ws, size_t ws_size,
                              hipStream_t stream) {
    const float* x = (const float*)d_in[0];
    const float* y = (const float*)d_in[1];
    float* ws = (float*)d_ws;

    const int N      = in_sizes[0] / CL_D;   // 200000
    const int ntiles = N / CL_TROWS;         // 12500 (N divisible by 16)

    cl_zero<<<(WS_ZERO_N + 255) / 256, 256, 0, stream>>>(ws, WS_ZERO_N);

    cl_sums<<<512, 512, 0, stream>>>(x, y, ws + WS_SUMS, ws + WS_COUNTS, ntiles);

    cl_center<<<(CL_C * CL_D + 255) / 256, 256, 0, stream>>>(
        ws + WS_SUMS, ws + WS_COUNTS, ws + WS_CENTER);

    cl_dist<<<1024, 256, 0, stream>>>(x, y, ws + WS_CENTER, ws + WS_DSUM, N);

    cl_final<<<1, 32, 0, stream>>>(ws + WS_CENTER, ws + WS_COUNTS, ws + WS_DSUM,
                                   (float*)d_out);
}